// GATConvLayer_3470333575820
// MI455X (gfx1250) — compile-verified
//
#include <hip/hip_runtime.h>
#include <math.h>

#define D 128

typedef __attribute__((ext_vector_type(2))) float v2f;
typedef __attribute__((ext_vector_type(8))) float v8f;

// ---------------------------------------------------------------------------
// Y[N,128] = X[N,128] @ W[128,128] + bias[128]   (fp32 WMMA 16x16x4)
// One wave computes one 16x16 output tile; 8 waves/block cover all 128 cols.
// ---------------------------------------------------------------------------
__global__ void __launch_bounds__(256) gemm_bias_f32_wmma(
    const float* __restrict__ X, const float* __restrict__ W,
    const float* __restrict__ Bv, float* __restrict__ Y, int Nrows)
{
    const int wave = threadIdx.x >> 5;
    const int lane = threadIdx.x & 31;
    const int half = lane >> 4;      // 0: lanes 0-15, 1: lanes 16-31
    const int lid  = lane & 15;
    const int m0   = blockIdx.x * 16;
    const int n0   = wave * 16;
    if (m0 >= Nrows) return;         // block-uniform

    int mrow = m0 + lid;
    if (mrow >= Nrows) mrow = Nrows - 1;   // clamp (stores guarded below)
    const float* __restrict__ xrow = X + (size_t)mrow * D;

    v8f acc = {};
    #pragma unroll
    for (int k0 = 0; k0 < D; k0 += 4) {
        const int k = k0 + 2 * half; // A 16x4 f32 layout: lanes0-15 K=0,1 ; lanes16-31 K=2,3
        v2f a, b;
        a.x = xrow[k];
        a.y = xrow[k + 1];
        b.x = W[(size_t)k * D + n0 + lid];        // B 4x16: row K across lanes
        b.y = W[(size_t)(k + 1) * D + n0 + lid];
        acc = __builtin_amdgcn_wmma_f32_16x16x4_f32(
            false, a, false, b, (short)0, acc, false, false);
    }

    const float bias = Bv[n0 + lid];
    #pragma unroll
    for (int r = 0; r < 8; ++r) {
        const int m = m0 + r + 8 * half;  // C/D layout: VGPR r -> M=r (lo) / M=r+8 (hi)
        if (m < Nrows)
            Y[(size_t)m * D + n0 + lid] = acc[r] + bias;
    }
}

// ---------------------------------------------------------------------------
// init: emax = -inf, denom = 0, agg = 0
// ---------------------------------------------------------------------------
__global__ void init_buffers(float* __restrict__ emax, float* __restrict__ denom,
                             float* __restrict__ agg, int Nn)
{
    const int idx = blockIdx.x * blockDim.x + threadIdx.x;
    if (idx < Nn) { emax[idx] = -INFINITY; denom[idx] = 0.0f; }
    if (idx < Nn * D) agg[idx] = 0.0f;
}

__device__ __forceinline__ void atomicMaxFloat(float* addr, float val)
{
    if (val >= 0.0f)
        atomicMax((int*)addr, __float_as_int(val));
    else
        atomicMin((unsigned int*)addr, __float_as_uint(val));
}

// ---------------------------------------------------------------------------
// Per-edge attention logit + segment max.  One wave per edge, 4 dims/lane.
// ---------------------------------------------------------------------------
__global__ void __launch_bounds__(256) edge_score(
    const int* __restrict__ src, const int* __restrict__ dst,
    const float* __restrict__ xl, const float* __restrict__ xr,
    const float* __restrict__ att,
    float* __restrict__ score, float* __restrict__ emax,
    int Etot, int Eorig)
{
    const int e    = (int)((blockIdx.x * (unsigned)blockDim.x + threadIdx.x) >> 5);
    const int lane = threadIdx.x & 31;
    if (e >= Etot) return;                       // wave-uniform
    const int j = (e < Eorig) ? src[e] : (e - Eorig);
    const int i = (e < Eorig) ? dst[e] : (e - Eorig);

    const float4* pl = (const float4*)(xl + (size_t)j * D);
    const float4* pr = (const float4*)(xr + (size_t)i * D);
    const float4* pa = (const float4*)att;
    float4 vl = pl[lane], vr = pr[lane], va = pa[lane];

    float m, s = 0.0f;
    m = vl.x + vr.x; s += (m > 0.0f ? m : 0.2f * m) * va.x;
    m = vl.y + vr.y; s += (m > 0.0f ? m : 0.2f * m) * va.y;
    m = vl.z + vr.z; s += (m > 0.0f ? m : 0.2f * m) * va.z;
    m = vl.w + vr.w; s += (m > 0.0f ? m : 0.2f * m) * va.w;

    #pragma unroll
    for (int off = 16; off > 0; off >>= 1)
        s += __shfl_xor(s, off, 32);

    if (lane == 0) {
        score[e] = s;
        atomicMaxFloat(&emax[i], s);
    }
}

// ---------------------------------------------------------------------------
// ex = exp(score - emax[dst]); denom[dst] += ex.  One thread per edge.
// ---------------------------------------------------------------------------
__global__ void edge_exp(const int* __restrict__ dst,
                         const float* __restrict__ score, const float* __restrict__ emax,
                         float* __restrict__ ex, float* __restrict__ denom,
                         int Etot, int Eorig)
{
    const int e = blockIdx.x * blockDim.x + threadIdx.x;
    if (e >= Etot) return;
    const int i = (e < Eorig) ? dst[e] : (e - Eorig);
    const float v = __expf(score[e] - emax[i]);
    ex[e] = v;
    atomicAdd(&denom[i], v);
}

// ---------------------------------------------------------------------------
// agg[dst] += (ex/denom[dst]) * xl[src].  One wave per edge, float atomics to L2.
// ---------------------------------------------------------------------------
__global__ void __launch_bounds__(256) edge_scatter(
    const int* __restrict__ src, const int* __restrict__ dst,
    const float* __restrict__ xl,
    const float* __restrict__ ex, const float* __restrict__ denom,
    float* __restrict__ agg, int Etot, int Eorig)
{
    const int e    = (int)((blockIdx.x * (unsigned)blockDim.x + threadIdx.x) >> 5);
    const int lane = threadIdx.x & 31;
    if (e >= Etot) return;                       // wave-uniform
    const int j = (e < Eorig) ? src[e] : (e - Eorig);
    const int i = (e < Eorig) ? dst[e] : (e - Eorig);
    const float alpha = ex[e] / (denom[i] + 1e-16f);

    const float4 v = ((const float4*)(xl + (size_t)j * D))[lane];
    float* po = agg + (size_t)i * D + lane * 4;
    atomicAdd(po + 0, alpha * v.x);
    atomicAdd(po + 1, alpha * v.y);
    atomicAdd(po + 2, alpha * v.z);
    atomicAdd(po + 3, alpha * v.w);
}

// ---------------------------------------------------------------------------
// out = relu(agg + bias + res)
// ---------------------------------------------------------------------------
__global__ void finalize(const float* __restrict__ agg, const float* __restrict__ res,
                         const float* __restrict__ bias, float* __restrict__ out, int Nn)
{
    const int idx = blockIdx.x * blockDim.x + threadIdx.x;
    if (idx >= Nn * D) return;
    const int d = idx & (D - 1);
    const float v = agg[idx] + bias[d] + res[idx];
    out[idx] = v > 0.0f ? v : 0.0f;
}

// ---------------------------------------------------------------------------
extern "C" void kernel_launch(void* const* d_in, const int* in_sizes, int n_in,
                              void* d_out, int out_size, void* d_ws, size_t ws_size,
                              hipStream_t stream)
{
    (void)n_in; (void)out_size; (void)ws_size;

    const float* x     = (const float*)d_in[0];
    const int*   ei    = (const int*)  d_in[1];
    const float* Wl    = (const float*)d_in[2];
    const float* bl    = (const float*)d_in[3];
    const float* Wr    = (const float*)d_in[4];
    const float* br    = (const float*)d_in[5];
    const float* att   = (const float*)d_in[6];
    const float* biasL = (const float*)d_in[7];
    const float* Wres  = (const float*)d_in[8];
    const float* bres  = (const float*)d_in[9];

    const int N = in_sizes[0] / D;
    const int E = in_sizes[1] / 2;
    const int L = in_sizes[2] / (D * D);
    const int Etot = E + N;

    const int* srcIdx = ei;       // edge_index[0]
    const int* dstIdx = ei + E;   // edge_index[1]

    // workspace layout (floats)
    float* ws = (float*)d_ws;
    const size_t nd = (size_t)N * D;
    float* xl    = ws;
    float* xr    = xl + nd;
    float* res   = xr + nd;
    float* agg   = res + nd;
    float* hbuf  = agg + nd;
    float* score = hbuf + nd;
    float* ex    = score + Etot;
    float* emax  = ex + Etot;
    float* denom = emax + N;

    const dim3 blk(256);
    const dim3 gemmGrid((N + 15) / 16);
    const dim3 ndGrid((N * D + 255) / 256);
    const dim3 edgeWaveGrid(((size_t)Etot * 32 + 255) / 256);
    const dim3 edgeGrid((Etot + 255) / 256);

    const float* h = x;
    for (int l = 0; l < L; ++l) {
        const float* Wl_ = Wl + (size_t)l * D * D;
        const float* Wr_ = Wr + (size_t)l * D * D;
        const float* bl_ = bl + (size_t)l * D;
        const float* br_ = br + (size_t)l * D;
        const float* at_ = att + (size_t)l * D;
        const float* bi_ = biasL + (size_t)l * D;

        gemm_bias_f32_wmma<<<gemmGrid, blk, 0, stream>>>(h, Wl_,  bl_,  xl,  N);
        gemm_bias_f32_wmma<<<gemmGrid, blk, 0, stream>>>(h, Wr_,  br_,  xr,  N);
        gemm_bias_f32_wmma<<<gemmGrid, blk, 0, stream>>>(h, Wres, bres, res, N);

        init_buffers<<<ndGrid, blk, 0, stream>>>(emax, denom, agg, N);
        edge_score  <<<edgeWaveGrid, blk, 0, stream>>>(srcIdx, dstIdx, xl, xr, at_,
                                                       score, emax, Etot, E);
        edge_exp    <<<edgeGrid, blk, 0, stream>>>(dstIdx, score, emax, ex, denom, Etot, E);
        edge_scatter<<<edgeWaveGrid, blk, 0, stream>>>(srcIdx, dstIdx, xl, ex, denom,
                                                       agg, Etot, E);

        float* out = (l == L - 1) ? (float*)d_out : hbuf;
        finalize<<<ndGrid, blk, 0, stream>>>(agg, res, bi_, out, N);
        h = hbuf;
    }
}